// DSSIMLoss_77360950935833
// MI455X (gfx1250) — compile-verified
//
#include <hip/hip_runtime.h>

typedef float v2f __attribute__((ext_vector_type(2)));
typedef float v8f __attribute__((ext_vector_type(8)));

#define KS    11
#define RAD   5
#define TILE  16
#define EXT   26      // TILE + 2*RAD
#define INW   28      // K padded to multiple of 4
#define INH   32      // rows padded so row-tile 1 A-reads stay in-bounds
#define NCH   7       // 28 / 4 K-chunks
#define WAVES 4
#define IMG   512
#define NPLANE 96     // 32 batches * 3 channels
#define NPIX  25165824.0f

// Separable normalized Gaussian, ksize=11 sigma=1.5 (sums to 1)
__constant__ float cG[KS] = {
    0.0010284f, 0.0075988f, 0.0360008f, 0.1093607f, 0.2130054f,
    0.2660117f,
    0.2130054f, 0.1093607f, 0.0360008f, 0.0075988f, 0.0010284f};

__global__ __launch_bounds__(128) void ssim_tile_kernel(
    const float* __restrict__ pred, const float* __restrict__ targ,
    float* __restrict__ partial)
{
    __shared__ float sP[WAVES][INH][INW];
    __shared__ float sT[WAVES][INH][INW];
    __shared__ float sM[WAVES][5][INH][TILE];   // horizontal-blur intermediates
    __shared__ float sRed[WAVES];

    const int tid  = threadIdx.x;
    const int wid  = tid >> 5;
    const int lane = tid & 31;

    const int plane  = blockIdx.z;
    const size_t poff = (size_t)plane * (IMG * IMG);
    const int base_r = blockIdx.y * TILE;
    const int base_c = (blockIdx.x * WAVES + wid) * TILE;

    // ---- Stage 26x26 halo region of pred/target into LDS (zero pad) ----
    for (int i = lane; i < INH * INW; i += 32) {
        int lr = i / INW, lc = i - lr * INW;
        int r = base_r - RAD + lr, c = base_c - RAD + lc;
        float p = 0.f, t = 0.f;
        if (lr < EXT && lc < EXT && r >= 0 && r < IMG && c >= 0 && c < IMG) {
            size_t g = poff + (size_t)r * IMG + c;
            p = pred[g];
            t = targ[g];
        }
        sP[wid][lr][lc] = p;
        sT[wid][lr][lc] = t;
    }
    __syncthreads();

    const int n    = lane & 15;           // M (A) / N (B,D) index
    const int koff = (lane >> 4) << 1;    // K sub-offset from lane half
    const int rofs = (lane >> 4) << 3;    // D-layout row offset

    // Per-lane band-matrix weights g[K - n]; identical for the horizontal
    // B-operand and the vertical A-operand (same lane->K/index mapping).
    v2f wcf[NCH];
#pragma unroll
    for (int k = 0; k < NCH; ++k) {
        int K0 = 4 * k + koff;
        int d0 = K0 - n, d1 = K0 + 1 - n;
        int c0 = d0 < 0 ? 0 : (d0 > 10 ? 10 : d0);
        int c1 = d1 < 0 ? 0 : (d1 > 10 ? 10 : d1);
        v2f w;
        w[0] = (d0 >= 0 && d0 < KS) ? cG[c0] : 0.f;
        w[1] = (d1 >= 0 && d1 < KS) ? cG[c1] : 0.f;
        wcf[k] = w;
    }

    const v8f vzero = {0.f, 0.f, 0.f, 0.f, 0.f, 0.f, 0.f, 0.f};

    // ---- Pass 1: horizontal blur via WMMA, 5 quantities, 2 row-tiles ----
#pragma unroll
    for (int T = 0; T < 2; ++T) {
        v8f acc[5];
#pragma unroll
        for (int q = 0; q < 5; ++q) acc[q] = vzero;

        const int row = 16 * T + n;       // A-matrix row for this lane
#pragma unroll
        for (int k = 0; k < NCH; ++k) {
            int c0 = 4 * k + koff;
            float p0 = sP[wid][row][c0], p1 = sP[wid][row][c0 + 1];
            float t0 = sT[wid][row][c0], t1 = sT[wid][row][c0 + 1];
            v2f Ap  = {p0, p1};
            v2f At  = {t0, t1};
            v2f App = {p0 * p0, p1 * p1};
            v2f Att = {t0 * t0, t1 * t1};
            v2f Apt = {p0 * t0, p1 * t1};
            acc[0] = __builtin_amdgcn_wmma_f32_16x16x4_f32(false, Ap,  false, wcf[k], (short)0, acc[0], false, false);
            acc[1] = __builtin_amdgcn_wmma_f32_16x16x4_f32(false, At,  false, wcf[k], (short)0, acc[1], false, false);
            acc[2] = __builtin_amdgcn_wmma_f32_16x16x4_f32(false, App, false, wcf[k], (short)0, acc[2], false, false);
            acc[3] = __builtin_amdgcn_wmma_f32_16x16x4_f32(false, Att, false, wcf[k], (short)0, acc[3], false, false);
            acc[4] = __builtin_amdgcn_wmma_f32_16x16x4_f32(false, Apt, false, wcf[k], (short)0, acc[4], false, false);
        }
        // D layout: vgpr i -> rows (i, i+8) per lane half; store to LDS
#pragma unroll
        for (int q = 0; q < 5; ++q)
#pragma unroll
            for (int i = 0; i < 8; ++i)
                sM[wid][q][16 * T + rofs + i][n] = acc[q][i];
    }
    __syncthreads();

    // ---- Pass 2: vertical blur via WMMA (constant band A, tmp as B) ----
    v8f outq[5];
#pragma unroll
    for (int q = 0; q < 5; ++q) outq[q] = vzero;

#pragma unroll
    for (int k = 0; k < NCH; ++k) {
        int r0 = 4 * k + koff;
#pragma unroll
        for (int q = 0; q < 5; ++q) {
            v2f b = {sM[wid][q][r0][n], sM[wid][q][r0 + 1][n]};
            outq[q] = __builtin_amdgcn_wmma_f32_16x16x4_f32(false, wcf[k], false, b, (short)0, outq[q], false, false);
        }
    }

    // ---- Elementwise SSIM + tile partial sum ----
    const float C1 = 0.0001f;   // 0.01^2
    const float C2 = 0.0009f;   // 0.03^2
    float lsum = 0.f;
#pragma unroll
    for (int i = 0; i < 8; ++i) {
        float mp  = outq[0][i], mt = outq[1][i];
        float epp = outq[2][i], ett = outq[3][i], ept = outq[4][i];
        float mp2 = mp * mp, mt2 = mt * mt, mpt = mp * mt;
        float sp2 = epp - mp2, st2 = ett - mt2, spt = ept - mpt;
        float num = (2.f * mpt + C1) * (2.f * spt + C2);
        float den = (mp2 + mt2 + C1) * (sp2 + st2 + C2);
        lsum += 1.f - num / den;
    }
#pragma unroll
    for (int o = 16; o > 0; o >>= 1) lsum += __shfl_xor(lsum, o, 32);
    if (lane == 0) sRed[wid] = lsum;
    __syncthreads();
    if (tid == 0) {
        float s = sRed[0] + sRed[1] + sRed[2] + sRed[3];
        int bidx = (blockIdx.z * gridDim.y + blockIdx.y) * gridDim.x + blockIdx.x;
        partial[bidx] = s;
    }
}

__global__ __launch_bounds__(256) void ssim_reduce_kernel(
    const float* __restrict__ partial, float* __restrict__ out,
    int nblk, float scale)
{
    __shared__ float s[8];
    float v = 0.f;
    for (int i = threadIdx.x; i < nblk; i += 256) v += partial[i];
#pragma unroll
    for (int o = 16; o > 0; o >>= 1) v += __shfl_xor(v, o, 32);
    if ((threadIdx.x & 31) == 0) s[threadIdx.x >> 5] = v;
    __syncthreads();
    if (threadIdx.x == 0) {
        float t = 0.f;
#pragma unroll
        for (int w = 0; w < 8; ++w) t += s[w];
        out[0] = t * scale;
    }
}

extern "C" void kernel_launch(void* const* d_in, const int* in_sizes, int n_in,
                              void* d_out, int out_size, void* d_ws, size_t ws_size,
                              hipStream_t stream) {
    const float* pred = (const float*)d_in[0];
    const float* targ = (const float*)d_in[1];
    float* out      = (float*)d_out;
    float* partial  = (float*)d_ws;

    dim3 grid(IMG / (TILE * WAVES), IMG / TILE, NPLANE);   // (8, 32, 96)
    dim3 block(32 * WAVES);                                // 4 wave32s
    ssim_tile_kernel<<<grid, block, 0, stream>>>(pred, targ, partial);

    const int nblk = (IMG / (TILE * WAVES)) * (IMG / TILE) * NPLANE;  // 24576
    ssim_reduce_kernel<<<1, 256, 0, stream>>>(partial, out, nblk, 1.0f / NPIX);
}